// RGAT_8564164788310
// MI455X (gfx1250) — compile-verified
//
#include <hip/hip_runtime.h>
#include <hip/hip_bf16.h>

typedef __attribute__((ext_vector_type(2))) float v2f;
typedef __attribute__((ext_vector_type(8))) float v8f;

#define NSLOPE 0.2f
#define CDIM   128

// ---------------------------------------------------------------------------
// Dense per-relation transform via V_WMMA_F32_16X16X4_F32, with fused q/k
// projections.  One wave per (relation, 16-node tile); 8 v8f accumulators
// cover the 128-wide output.
// ---------------------------------------------------------------------------
__global__ __launch_bounds__(256)
void rgat_gemm_wmma(const float* __restrict__ x,      // [N,128]
                    const float* __restrict__ W,      // [R,128,128]
                    const float* __restrict__ qv,     // [128]
                    const float* __restrict__ kv,     // [128]
                    float* __restrict__ xr,           // [R,N,128]
                    float* __restrict__ qproj,        // [R,N]
                    float* __restrict__ kproj,        // [R,N]
                    int nnodes, int ntiles, int nrel)
{
    const int wave = (blockIdx.x * blockDim.x + threadIdx.x) >> 5;
    const int lane = threadIdx.x & 31;
    if (wave >= nrel * ntiles) return;
    const int r  = wave / ntiles;
    const int n0 = (wave - r * ntiles) * 16;
    const int m  = lane & 15;   // row (A) / col (B,C)
    const int hi = lane >> 4;   // K-pair select / C row-half select

    const float* __restrict__ xrow = x + (size_t)(n0 + m) * CDIM;
    const float* __restrict__ Wr   = W + (size_t)r * CDIM * CDIM;

    v8f acc[8] = {};

    for (int k0 = 0; k0 < CDIM; k0 += 4) {
        const int kk = k0 + 2 * hi;
        v2f a;
        a.x = xrow[kk];
        a.y = xrow[kk + 1];
        const float* __restrict__ Wk0 = Wr + (size_t)kk * CDIM;
        const float* __restrict__ Wk1 = Wk0 + CDIM;
        #pragma unroll
        for (int t = 0; t < 8; ++t) {
            v2f b;
            b.x = Wk0[t * 16 + m];
            b.y = Wk1[t * 16 + m];
            acc[t] = __builtin_amdgcn_wmma_f32_16x16x4_f32(
                false, a, false, b, (short)0, acc[t], false, false);
        }
    }

    // Store xr tile.  C layout: VGPR j -> row j (lanes 0-15) / row j+8 (16-31).
    float* __restrict__ xout = xr + ((size_t)r * nnodes + n0) * CDIM;
    #pragma unroll
    for (int t = 0; t < 8; ++t)
        #pragma unroll
        for (int j = 0; j < 8; ++j)
            xout[(size_t)(j + 8 * hi) * CDIM + t * 16 + m] = acc[t][j];

    // Fused q/k projections: per-lane partial dot over held columns, then
    // butterfly-reduce across the 16 lanes of each half (masks < 16 keep the
    // two halves, which own disjoint row sets, separate).
    float qp[8], kp[8];
    #pragma unroll
    for (int j = 0; j < 8; ++j) { qp[j] = 0.f; kp[j] = 0.f; }
    #pragma unroll
    for (int t = 0; t < 8; ++t) {
        const float qc = qv[t * 16 + m];
        const float kc = kv[t * 16 + m];
        #pragma unroll
        for (int j = 0; j < 8; ++j) {
            qp[j] += acc[t][j] * qc;
            kp[j] += acc[t][j] * kc;
        }
    }
    #pragma unroll
    for (int off = 1; off < 16; off <<= 1) {
        #pragma unroll
        for (int j = 0; j < 8; ++j) {
            qp[j] += __shfl_xor(qp[j], off, 32);
            kp[j] += __shfl_xor(kp[j], off, 32);
        }
    }
    if (m == 0) {
        #pragma unroll
        for (int j = 0; j < 8; ++j) {
            const int row = n0 + j + 8 * hi;
            qproj[(size_t)r * nnodes + row] = qp[j];
            kproj[(size_t)r * nnodes + row] = kp[j];
        }
    }
}

// ---------------------------------------------------------------------------
// Pass 1: alpha = leaky_relu(q[dst] + k[src]); segment-max via ordered-key
// uint atomicMax (monotone float<->uint mapping).
// ---------------------------------------------------------------------------
__device__ __forceinline__ unsigned f32_key(float f) {
    unsigned b = __float_as_uint(f);
    return (b & 0x80000000u) ? ~b : (b | 0x80000000u);
}
__device__ __forceinline__ float key_f32(unsigned k) {
    return __uint_as_float((k & 0x80000000u) ? (k ^ 0x80000000u) : ~k);
}

__global__ __launch_bounds__(256)
void edge_alpha_max(const int* __restrict__ ei, const int* __restrict__ et,
                    const float* __restrict__ qproj, const float* __restrict__ kproj,
                    float* __restrict__ alpha, unsigned* __restrict__ amaxk,
                    int nedges, int nnodes)
{
    const int e = blockIdx.x * blockDim.x + threadIdx.x;
    if (e >= nedges) return;
    const int s = ei[e];
    const int d = ei[nedges + e];
    const int r = et[e];
    float a = qproj[(size_t)r * nnodes + d] + kproj[(size_t)r * nnodes + s];
    a = (a > 0.f) ? a : NSLOPE * a;
    alpha[e] = a;
    atomicMax(&amaxk[d], f32_key(a));
}

// ---------------------------------------------------------------------------
// Pass 2: ex = exp(alpha - amax[dst]) (in place); denom[dst] += ex.
// ---------------------------------------------------------------------------
__global__ __launch_bounds__(256)
void edge_exp_sum(const int* __restrict__ ei, float* __restrict__ alpha,
                  const unsigned* __restrict__ amaxk, float* __restrict__ denom,
                  int nedges)
{
    const int e = blockIdx.x * blockDim.x + threadIdx.x;
    if (e >= nedges) return;
    const int d = ei[nedges + e];
    const float v = __expf(alpha[e] - key_f32(amaxk[d]));
    alpha[e] = v;                      // reuse buffer: alpha -> ex
    atomicAdd(&denom[d], v);
}

// ---------------------------------------------------------------------------
// Pass 3: one wave per edge, float4 per lane (128 floats / 32 lanes):
// out[dst] += (ex/denom[dst]) * xr[etype, src].  Accumulator (25.6 MB) is
// L2-resident, so atomics never hit HBM.
// ---------------------------------------------------------------------------
__global__ __launch_bounds__(256)
void edge_scatter(const int* __restrict__ ei, const int* __restrict__ et,
                  const float* __restrict__ ex, const float* __restrict__ denom,
                  const float* __restrict__ xr, float* __restrict__ out,
                  int nedges, int nnodes)
{
    const long long gid = (long long)blockIdx.x * blockDim.x + threadIdx.x;
    const int e    = (int)(gid >> 5);
    const int lane = (int)(gid & 31);
    if (e >= nedges) return;
    const int s = ei[e];
    const int d = ei[nedges + e];
    const int r = et[e];
    const float w = ex[e] / (denom[d] + 1e-16f);
    const float4 v = ((const float4*)(xr + ((size_t)r * nnodes + s) * CDIM))[lane];
    float* dp = out + (size_t)d * CDIM + lane * 4;
    atomicAdd(dp + 0, w * v.x);
    atomicAdd(dp + 1, w * v.y);
    atomicAdd(dp + 2, w * v.z);
    atomicAdd(dp + 3, w * v.w);
}

// ---------------------------------------------------------------------------
// Epilogue: h = (relu?) (h + bias[d])
// ---------------------------------------------------------------------------
__global__ __launch_bounds__(256)
void epilogue(float* __restrict__ h, const float* __restrict__ b,
              size_t total, int do_relu)
{
    const size_t i = (size_t)blockIdx.x * blockDim.x + threadIdx.x;
    if (i >= total) return;
    float v = h[i] + b[i & (CDIM - 1)];
    h[i] = do_relu ? fmaxf(v, 0.f) : v;
}

// ---------------------------------------------------------------------------
extern "C" void kernel_launch(void* const* d_in, const int* in_sizes, int n_in,
                              void* d_out, int out_size, void* d_ws, size_t ws_size,
                              hipStream_t stream)
{
    const float* x  = (const float*)d_in[0];
    const int*   ei = (const int*)  d_in[1];
    const int*   et = (const int*)  d_in[2];

    const int nnodes = in_sizes[0] / CDIM;
    const int nedges = in_sizes[2];
    const int nrel   = in_sizes[3] / (CDIM * CDIM);
    const int ntiles = (nnodes + 15) / 16;

    // ----- workspace carve (256B aligned) -----
    char* p = (char*)d_ws;
    auto carve = [&](size_t bytes) -> void* {
        void* q = (void*)p;
        p += (bytes + 255) & ~(size_t)255;
        return q;
    };
    float*    xr    = (float*)   carve((size_t)nrel * nnodes * CDIM * 4);
    float*    qproj = (float*)   carve((size_t)nrel * nnodes * 4);
    float*    kproj = (float*)   carve((size_t)nrel * nnodes * 4);
    float*    alpha = (float*)   carve((size_t)nedges * 4);
    unsigned* amaxk = (unsigned*)carve((size_t)nnodes * 4);
    float*    denom = (float*)   carve((size_t)nnodes * 4);
    float*    h0    = (float*)   carve((size_t)nnodes * CDIM * 4);
    float*    h1    = (float*)   carve((size_t)nnodes * CDIM * 4);
    if ((size_t)(p - (char*)d_ws) > ws_size) return;  // insufficient scratch

    const float* in_ptr = x;
    for (int layer = 0; layer < 3; ++layer) {
        const float* Wl = (const float*)d_in[3 + 4 * layer];
        const float* ql = (const float*)d_in[4 + 4 * layer];
        const float* kl = (const float*)d_in[5 + 4 * layer];
        const float* bl = (const float*)d_in[6 + 4 * layer];
        float* out_ptr = (layer == 0) ? h0 : (layer == 1) ? h1 : (float*)d_out;

        hipMemsetAsync(amaxk,   0, (size_t)nnodes * 4, stream);
        hipMemsetAsync(denom,   0, (size_t)nnodes * 4, stream);
        hipMemsetAsync(out_ptr, 0, (size_t)nnodes * CDIM * 4, stream);

        {   // dense transform + fused projections (WMMA)
            const int nwaves = nrel * ntiles;
            dim3 b(256), g((unsigned)((nwaves * 32 + 255) / 256));
            rgat_gemm_wmma<<<g, b, 0, stream>>>(in_ptr, Wl, ql, kl,
                                                xr, qproj, kproj,
                                                nnodes, ntiles, nrel);
        }
        {   dim3 b(256), g((unsigned)((nedges + 255) / 256));
            edge_alpha_max<<<g, b, 0, stream>>>(ei, et, qproj, kproj,
                                                alpha, amaxk, nedges, nnodes);
        }
        {   dim3 b(256), g((unsigned)((nedges + 255) / 256));
            edge_exp_sum<<<g, b, 0, stream>>>(ei, alpha, amaxk, denom, nedges);
        }
        {   const long long th = (long long)nedges * 32;
            dim3 b(256), g((unsigned)((th + 255) / 256));
            edge_scatter<<<g, b, 0, stream>>>(ei, et, alpha, denom, xr,
                                              out_ptr, nedges, nnodes);
        }
        {   const size_t tot = (size_t)nnodes * CDIM;
            dim3 b(256), g((unsigned)((tot + 255) / 256));
            epilogue<<<g, b, 0, stream>>>(out_ptr, bl, tot, layer < 2 ? 1 : 0);
        }
        in_ptr = out_ptr;
    }
}